// PointPillarsLoss_84456236908807
// MI455X (gfx1250) — compile-verified
//
#include <hip/hip_runtime.h>
#include <hip/hip_bf16.h>

typedef __attribute__((ext_vector_type(2))) float v2f;
typedef __attribute__((ext_vector_type(8))) float v8f;

#define B_    16
#define H_    250
#define W_    500
#define N_    64
#define HW_   (H_ * W_)            // 125000
#define BHW_  (B_ * HW_)           // 2000000
#define NBLKX ((HW_ + 255) / 256)  // 489
#define NPART (B_ * NBLKX)         // 7824

// d_ws layout (bytes):
//   [0, BHW*4)                : boxgrid  u32  (0 = no box, else boxIdxInBatch+1)
//   [BHW*4, BHW*5)            : igngrid  u8
//   [10'000'000, +32768)      : boxtab   f32 [B*N][8] = {dx,dy,z,logl,logw,logh,sinr,dirbin}
//   [10'032'768, +NPART*5*4)  : partials f32 [NPART][5] = {cls_sum,valid0,pos,reg_sum,dir_sum}

__global__ void ppl_zero(unsigned* __restrict__ p, int n) {
  int i = blockIdx.x * blockDim.x + threadIdx.x;
  int stride = gridDim.x * blockDim.x;
  for (; i < n; i += stride) p[i] = 0u;
}

__global__ void ppl_boxes(const float* __restrict__ gt,
                          unsigned* __restrict__ boxgrid,
                          unsigned char* __restrict__ igngrid,
                          float* __restrict__ boxtab) {
  int idx = blockIdx.x * blockDim.x + threadIdx.x;
  if (idx >= B_ * N_) return;
  int b = idx / N_, n = idx % N_;
  const float* g = gt + (size_t)idx * 8;
  float x = g[0], y = g[1], z = g[2], l = g[3], w = g[4], h = g[5], rot = g[6], cid = g[7];
  bool valid = (cid == 0.0f) && (x >= 0.0f) && (x < 200.0f) && (y >= -50.0f) && (y < 50.0f);
  int gx = (int)floorf(x / 0.4f);
  int gy = (int)floorf((y + 50.0f) / 0.4f);
  valid = valid && (gx >= 0) && (gx < W_) && (gy >= 0) && (gy < H_);

  float cx = ((float)gx + 0.5f) * 0.4f;              // X_MIN = 0
  float cy = -50.0f + ((float)gy + 0.5f) * 0.4f;
  float t0 = (x - cx) / 0.4f;
  float t1 = (y - cy) / 0.4f;
  float t3 = logf(fmaxf(l, 1e-3f));
  float t4 = logf(fmaxf(w, 1e-3f));
  float t5 = logf(fmaxf(h, 1e-3f));
  float t6 = sinf(rot);
  float t7 = (cosf(rot) >= 0.0f) ? 0.0f : 1.0f;

  float4* bt = (float4*)(boxtab + (size_t)idx * 8);
  bt[0] = make_float4(t0, t1, z, t3);
  bt[1] = make_float4(t4, t5, t6, t7);

  if (valid) {
    int flat = (b * H_ + gy) * W_ + gx;
    atomicMax(&boxgrid[flat], (unsigned)(n + 1));
#pragma unroll
    for (int oy = -1; oy <= 1; ++oy)
#pragma unroll
      for (int ox = -1; ox <= 1; ++ox) {
        int gy2 = gy + oy, gx2 = gx + ox;
        if (gy2 >= 0 && gy2 < H_ && gx2 >= 0 && gx2 < W_)
          igngrid[(b * H_ + gy2) * W_ + gx2] = 1;
      }
  }
}

__device__ __forceinline__ float bce_logits(float x, float t, float e) {
  // e = exp(-|x|), precomputed
  return fmaxf(x, 0.0f) - x * t + log1pf(e);
}

__global__ void __launch_bounds__(256)
ppl_main(const float* __restrict__ cls, const float* __restrict__ reg,
         const float* __restrict__ dir, const unsigned* __restrict__ boxgrid,
         const unsigned char* __restrict__ igngrid, const float* __restrict__ boxtab,
         float* __restrict__ partial) {
  int b = blockIdx.y;
  int hw = blockIdx.x * 256 + threadIdx.x;
  float cls_sum = 0.0f, v0 = 0.0f, pos_cnt = 0.0f, reg_sum = 0.0f, dir_sum = 0.0f;

  if (hw < HW_) {
    int cell = b * HW_ + hw;
    unsigned bv = boxgrid[cell];
    bool pos = (bv != 0u);
    bool ign = (igngrid[cell] != 0);

    const float* bt = nullptr;
    if (pos) {
      bt = boxtab + ((size_t)(b * N_) + (int)bv - 1) * 8;
      // gfx1250: global_prefetch_b8 the strided gather lines while cls math runs
      __builtin_prefetch(bt, 0, 0);
#pragma unroll
      for (int k = 0; k < 7; ++k)
        __builtin_prefetch(&reg[(size_t)(b * 7 + k) * HW_ + hw], 0, 0);
#pragma unroll
      for (int k = 0; k < 2; ++k)
        __builtin_prefetch(&dir[(size_t)(b * 2 + k) * HW_ + hw], 0, 0);
    }

    // ---- focal loss: channel 0 (target = pos, valid = (~ign)|pos) ----
    {
      float x = cls[(size_t)(b * 3 + 0) * HW_ + hw];
      float tt = pos ? 1.0f : 0.0f;
      bool valid = (!ign) || pos;
      float e = expf(-fabsf(x));
      float p = (x >= 0.0f) ? 1.0f / (1.0f + e) : e / (1.0f + e);
      float pt = p * tt + (1.0f - p) * (1.0f - tt);
      float at = 0.25f * tt + 0.75f * (1.0f - tt);
      float om = 1.0f - pt;
      if (valid) {
        cls_sum += at * om * om * bce_logits(x, tt, e);
        v0 += 1.0f;
      }
    }
    // ---- channels 1,2: target 0, always valid ----
#pragma unroll
    for (int c = 1; c < 3; ++c) {
      float x = cls[(size_t)(b * 3 + c) * HW_ + hw];
      float e = expf(-fabsf(x));
      float p = (x >= 0.0f) ? 1.0f / (1.0f + e) : e / (1.0f + e);
      cls_sum += 0.75f * p * p * bce_logits(x, 0.0f, e);
    }

    // ---- reg smooth-L1 + dir BCE only on positive cells (~300 of 2M) ----
    if (pos) {
      pos_cnt = 1.0f;
#pragma unroll
      for (int k = 0; k < 7; ++k) {
        float r = reg[(size_t)(b * 7 + k) * HW_ + hw];
        float d = fabsf(r - bt[k]);
        reg_sum += (d < 1.0f) ? 0.5f * d * d : d - 0.5f;
      }
      float bin = bt[7];
#pragma unroll
      for (int k = 0; k < 2; ++k) {
        float x = dir[(size_t)(b * 2 + k) * HW_ + hw];
        float tt = ((float)k == bin) ? 1.0f : 0.0f;
        float e = expf(-fabsf(x));
        dir_sum += bce_logits(x, tt, e);
      }
    }
  }

  // ---- deterministic block reduction (wave32 shuffles + LDS) ----
  __shared__ float smem[8 * 5];
  float vals[5] = {cls_sum, v0, pos_cnt, reg_sum, dir_sum};
  int lane = threadIdx.x & 31;
  int wave = threadIdx.x >> 5;
#pragma unroll
  for (int j = 0; j < 5; ++j) {
    float v = vals[j];
#pragma unroll
    for (int off = 16; off > 0; off >>= 1) v += __shfl_down(v, off);
    if (lane == 0) smem[wave * 5 + j] = v;
  }
  __syncthreads();
  if (threadIdx.x == 0) {
    int pidx = blockIdx.y * gridDim.x + blockIdx.x;
#pragma unroll
    for (int j = 0; j < 5; ++j) {
      float s = 0.0f;
#pragma unroll
      for (int w2 = 0; w2 < 8; ++w2) s += smem[w2 * 5 + j];
      partial[(size_t)pidx * 5 + j] = s;
    }
  }
}

__global__ void __launch_bounds__(256)
ppl_final(const float* __restrict__ partial, float* __restrict__ out) {
  float s[5] = {0.f, 0.f, 0.f, 0.f, 0.f};
  for (int i = threadIdx.x; i < NPART; i += 256) {
#pragma unroll
    for (int j = 0; j < 5; ++j) s[j] += partial[(size_t)i * 5 + j];
  }

  // Cross-lane sum via V_WMMA_F32_16X16X4_F32 with all-ones B:
  // A[m] = [s_m, 0, s_(m+16), 0] -> D[m][*] = s_m + s_(m+16); sum lane's 8
  // D VGPRs = half-sum, shfl_xor(16) completes the 32-lane total.
  v2f bones;
  bones.x = 1.0f;
  bones.y = 1.0f;
  float wsum[5];
#pragma unroll
  for (int j = 0; j < 5; ++j) {
    v2f a;
    a.x = s[j];
    a.y = 0.0f;
    v8f c = {};
    c = __builtin_amdgcn_wmma_f32_16x16x4_f32(false, a, false, bones,
                                              (short)0, c, false, false);
    float t = c[0] + c[1] + c[2] + c[3] + c[4] + c[5] + c[6] + c[7];
    t += __shfl_xor(t, 16);
    wsum[j] = t;
  }

  __shared__ float smem[8 * 5];
  int lane = threadIdx.x & 31;
  int wave = threadIdx.x >> 5;
  if (lane == 0) {
#pragma unroll
    for (int j = 0; j < 5; ++j) smem[wave * 5 + j] = wsum[j];
  }
  __syncthreads();
  if (threadIdx.x == 0) {
    float tot[5] = {0.f, 0.f, 0.f, 0.f, 0.f};
#pragma unroll
    for (int w2 = 0; w2 < 8; ++w2)
#pragma unroll
      for (int j = 0; j < 5; ++j) tot[j] += smem[w2 * 5 + j];

    float vm = 2.0f * (float)BHW_ + tot[1];
    float cls_loss = tot[0] / fmaxf(vm, 1.0f);
    float reg_loss = tot[3] / fmaxf(7.0f * tot[2], 1.0f);
    float dir_loss = tot[4] / fmaxf(2.0f * tot[2], 1.0f);
    float total = cls_loss + 2.0f * reg_loss + 0.2f * dir_loss;
    out[0] = total;
    out[1] = cls_loss;
    out[2] = reg_loss;
    out[3] = dir_loss;
  }
}

extern "C" void kernel_launch(void* const* d_in, const int* in_sizes, int n_in,
                              void* d_out, int out_size, void* d_ws, size_t ws_size,
                              hipStream_t stream) {
  const float* cls = (const float*)d_in[0];  // [16,3,250,500]
  const float* reg = (const float*)d_in[1];  // [16,7,250,500]
  const float* dir = (const float*)d_in[2];  // [16,2,250,500]
  const float* gt  = (const float*)d_in[3];  // [16,64,8]
  (void)in_sizes; (void)n_in; (void)out_size; (void)ws_size;

  unsigned char* ws = (unsigned char*)d_ws;
  unsigned* boxgrid      = (unsigned*)ws;
  unsigned char* igngrid = ws + (size_t)BHW_ * 4;
  float* boxtab          = (float*)(ws + 10000000);  // 16B aligned
  float* partial         = (float*)(ws + 10032768);  // 16B aligned

  // zero boxgrid (8MB) + igngrid (2MB) = 2.5M dwords, every call
  ppl_zero<<<dim3(4096), dim3(256), 0, stream>>>((unsigned*)d_ws, (BHW_ * 5) / 4);
  ppl_boxes<<<dim3((B_ * N_ + 255) / 256), dim3(256), 0, stream>>>(gt, boxgrid, igngrid, boxtab);
  ppl_main<<<dim3(NBLKX, B_), dim3(256), 0, stream>>>(cls, reg, dir, boxgrid, igngrid,
                                                      boxtab, partial);
  ppl_final<<<dim3(1), dim3(256), 0, stream>>>(partial, (float*)d_out);
}